// PointBasedTransform_15032385536192
// MI455X (gfx1250) — compile-verified
//
#include <hip/hip_runtime.h>
#include <hip/hip_bf16.h>
#include <stdint.h>

// ---------------- static config (mirrors reference) ----------------
#define NPTS      3145728u          // N
#define GDIM      128               // G
#define NUM_VOX   2097152u          // G^3
#define NV1       (NUM_VOX + 1u)    // +1 sentinel bucket for OOB
#define KCAP      16u
#define D_DESC    24
#define M_RES     0.04f
#define HALF_CUBE 2.56f             // CUBE/2

// d_out layout (float32 elements; int32 regions share 4-byte cells)
#define PTS_OFF   0u
#define CONF_OFF  (3u * NPTS)
#define DESC_OFF  (4u * NPTS)
#define DC_OFF    (28u * NPTS)
#define FLOATS_END (29u * NPTS)                    // 91,226,112
#define GRID_ELEMS (NUM_VOX * KCAP)                // 33,554,432
#define INIT_TOTAL (FLOATS_END + GRID_ELEMS)       // 124,780,544

#define SCAN_TILE 4096u
#define COUNTS_PAD (513u * SCAN_TILE)              // 2,101,248 (>= NV1, tile multiple)
#define RADIX_BLOCKS (NPTS / 256u)                 // 12288, N % 256 == 0

typedef __attribute__((ext_vector_type(2))) float v2f;
typedef __attribute__((ext_vector_type(8))) float v8f;

// ---------------- init ----------------
__global__ void init_out_kernel(uint32_t* outw) {
  uint32_t stride = gridDim.x * blockDim.x;
  for (uint32_t i = blockIdx.x * blockDim.x + threadIdx.x; i < INIT_TOTAL; i += stride)
    outw[i] = (i < FLOATS_END) ? 0u : 0xFFFFFFFFu;   // zeros for floats, -1 for grid
}

__global__ void zero_u32_kernel(uint32_t* p, uint32_t n) {
  uint32_t i = blockIdx.x * blockDim.x + threadIdx.x;
  if (i < n) p[i] = 0u;
}

// ---------------- pass 1: voxel id + counts ----------------
__global__ void vid_count_kernel(const float* __restrict__ pts,
                                 const float* __restrict__ center,
                                 uint32_t* __restrict__ vidArr,
                                 uint32_t* __restrict__ counts) {
  uint32_t i = blockIdx.x * blockDim.x + threadIdx.x;
  if (i >= NPTS) return;
  float mcx = center[0] - HALF_CUBE;
  float mcy = center[1] - HALF_CUBE;
  float mcz = center[2] - HALF_CUBE;
  float x = pts[3u * i + 0u], y = pts[3u * i + 1u], z = pts[3u * i + 2u];
  int ix = (int)((x - mcx) / M_RES);   // trunc-toward-zero matches astype(int32)
  int iy = (int)((y - mcy) / M_RES);
  int iz = (int)((z - mcz) / M_RES);
  bool inb = (ix >= 0) && (ix < GDIM) && (iy >= 0) && (iy < GDIM) &&
             (iz >= 0) && (iz < GDIM);
  uint32_t vid = inb ? (uint32_t)(ix * GDIM * GDIM + iy * GDIM + iz) : NUM_VOX;
  vidArr[i] = vid;
  atomicAdd(&counts[vid], 1u);
}

// ---------------- exclusive scan (WMMA f32 16x16x4, exact: all sums < 2^24) --
// PRECONDITION: input readable (zero-padded) up to gridDim.x * SCAN_TILE.
__global__ void scanA_kernel(const uint32_t* __restrict__ in,
                             uint32_t* __restrict__ partials) {
  __shared__ uint32_t red[256];
  const uint4* in4 = (const uint4*)(in + blockIdx.x * SCAN_TILE) + threadIdx.x * 4u;
  uint32_t s = 0u;
  #pragma unroll
  for (uint32_t k = 0; k < 4u; ++k) {
    uint4 u = in4[k];
    s += u.x + u.y + u.z + u.w;
  }
  red[threadIdx.x] = s;
  __syncthreads();
  for (uint32_t off = 128u; off > 0u; off >>= 1) {
    if (threadIdx.x < off) red[threadIdx.x] += red[threadIdx.x + off];
    __syncthreads();
  }
  if (threadIdx.x == 0) partials[blockIdx.x] = red[0];
}

__global__ void scanB_kernel(uint32_t* partials, uint32_t nb) { // nb <= 1024
  __shared__ uint32_t s[1024];
  uint32_t t = threadIdx.x;
  uint32_t v = (t < nb) ? partials[t] : 0u;
  s[t] = v;
  __syncthreads();
  for (uint32_t off = 1u; off < 1024u; off <<= 1) {
    uint32_t x = (t >= off) ? s[t - off] : 0u;
    __syncthreads();
    s[t] += x;
    __syncthreads();
  }
  if (t < nb) partials[t] = s[t] - v;   // exclusive
}

// Per-wave scan of 256 elements as D = L(16x16, lower-tri ones) * X(16x16),
// built from four chained V_WMMA_F32_16X16X4_F32. Column n of X holds
// elements [16n,16n+16); D[m][n] = inclusive prefix within the column.
// Layouts per CDNA5 ISA 7.12.2:
//   A 16x4 f32 : lane<16 -> M=lane, K={c*4+j}; lane>=16 -> M=lane-16, K={c*4+j+2}
//   B 4x16 f32 : lane<16 -> K=c*4+j,   N=lane; lane>=16 -> K=c*4+j+2, N=lane-16
//   C/D 16x16  : lane<16 -> (M=v,  N=lane); lane>=16 -> (M=v+8, N=lane-16)
// B loads: per chunk c the two dwords are CONSECUTIVE -> one b64 load each.
__global__ void scanC_kernel(const uint32_t* __restrict__ in, uint32_t n,
                             const uint32_t* __restrict__ partials,
                             uint32_t* __restrict__ out,
                             uint32_t* __restrict__ out2) {
  __shared__ float wtot[8];
  const uint32_t wave = threadIdx.x >> 5;   // 8 waves of 32
  const uint32_t lane = threadIdx.x & 31u;
  const uint32_t half = lane >> 4;
  const uint32_t col  = lane & 15u;
  const uint32_t blockBase = blockIdx.x * SCAN_TILE;
  const float blockOff = (float)partials[blockIdx.x];

  v2f a[4];
  #pragma unroll
  for (int c = 0; c < 4; ++c)
    #pragma unroll
    for (int j = 0; j < 2; ++j)
      a[c][j] = ((uint32_t)(4 * c + j) + 2u * half <= col) ? 1.0f : 0.0f;

  float vals[2][8];
  float chunkPref = 0.0f;
  #pragma unroll
  for (int ch = 0; ch < 2; ++ch) {
    uint32_t base = blockBase + wave * 512u + (uint32_t)ch * 256u;
    // unconditional 64-bit loads (input is tile-padded): 4 per chunk
    const uint2* bp = (const uint2*)(in + base + 16u * col + 2u * half);
    v8f acc = {0.0f, 0.0f, 0.0f, 0.0f, 0.0f, 0.0f, 0.0f, 0.0f};
    #pragma unroll
    for (int c = 0; c < 4; ++c) {
      uint2 u = bp[2 * c];                 // dwords {4c+2h, 4c+2h+1}
      v2f b;
      b[0] = (float)u.x;
      b[1] = (float)u.y;
      acc = __builtin_amdgcn_wmma_f32_16x16x4_f32(false, a[c], false, b,
                                                  (short)0, acc, false, false);
    }
    float s7 = acc[7];                     // lanes>=16 hold column totals S[15][n]
    float pref = 0.0f, tot = 0.0f;
    #pragma unroll
    for (int q = 0; q < 16; ++q) {
      float tq = __shfl(s7, 16 + q, 32);
      tot += tq;
      if ((uint32_t)q < col) pref += tq;
    }
    float e0 = half ? __shfl(s7, (int)col, 32) : 0.0f;  // S[7][col] for rows 8..15
    #pragma unroll
    for (int v = 0; v < 8; ++v) {
      float ex = (v == 0) ? e0 : acc[v - 1];   // exclusive within column
      vals[ch][v] = ex + pref + chunkPref;
    }
    chunkPref += tot;
  }
  if (lane == 0) wtot[wave] = chunkPref;
  __syncthreads();
  float woff = blockOff;
  for (uint32_t w = 0; w < wave; ++w) woff += wtot[w];
  #pragma unroll
  for (int ch = 0; ch < 2; ++ch) {
    uint32_t base = blockBase + wave * 512u + (uint32_t)ch * 256u;
    #pragma unroll
    for (int v = 0; v < 8; ++v) {
      uint32_t m = (uint32_t)v + 8u * half;
      uint32_t idx = base + 16u * col + m;
      if (idx < n) {
        uint32_t r = (uint32_t)(vals[ch][v] + woff + 0.5f);
        out[idx] = r;
        if (out2) out2[idx] = r;
      }
    }
  }
}

static void run_scan(const uint32_t* in, uint32_t n, uint32_t* out, uint32_t* out2,
                     uint32_t* partials, hipStream_t stream) {
  uint32_t nb = (n + SCAN_TILE - 1u) / SCAN_TILE;   // <= 1024 for all uses here
  scanA_kernel<<<nb, 256, 0, stream>>>(in, partials);
  scanB_kernel<<<1, 1024, 0, stream>>>(partials, nb);
  scanC_kernel<<<nb, 256, 0, stream>>>(in, n, partials, out, out2);
}

// ---------------- pass 2: bucket scatter ----------------
__global__ void scatter_kernel(const uint32_t* __restrict__ vidArr,
                               const float* __restrict__ conf,
                               uint32_t* __restrict__ cursor,
                               uint32_t* __restrict__ bPid,
                               float* __restrict__ bConf) {
  uint32_t i = blockIdx.x * blockDim.x + threadIdx.x;
  if (i >= NPTS) return;
  uint32_t vid = vidArr[i];
  uint32_t slot = atomicAdd(&cursor[vid], 1u);
  bPid[slot] = i;
  bConf[slot] = conf[i];
}

// ---------------- pass 3: per-voxel top-K selection (total order -> determinate)
__global__ void select_kernel(const uint32_t* __restrict__ counts,
                              const uint32_t* __restrict__ offsets,
                              uint32_t* __restrict__ bPid,
                              float* __restrict__ bConf,
                              int* __restrict__ voxcnt) {
  uint32_t v = blockIdx.x * blockDim.x + threadIdx.x;
  if (v >= NUM_VOX) return;
  uint32_t cnt = counts[v];
  uint32_t base = offsets[v];
  uint32_t kept = cnt < KCAP ? cnt : KCAP;
  for (uint32_t j = 0; j < kept; ++j) {          // partial selection sort:
    uint32_t best = j;                           // conf desc, tie: pid asc
    float bc = bConf[base + j];
    uint32_t bp = bPid[base + j];
    for (uint32_t q = j + 1u; q < cnt; ++q) {
      float c = bConf[base + q];
      uint32_t p = bPid[base + q];
      if (c > bc || (c == bc && p < bp)) { best = q; bc = c; bp = p; }
    }
    if (best != j) {
      bConf[base + best] = bConf[base + j]; bPid[base + best] = bPid[base + j];
      bConf[base + j] = bc; bPid[base + j] = bp;
    }
  }
  voxcnt[v] = (int)kept;
}

// ---------------- value radix-sort of conf (for conf_sorted faithful bug) ----
__global__ void make_keys_kernel(const float* __restrict__ conf,
                                 uint32_t* __restrict__ keys) {
  uint32_t i = blockIdx.x * blockDim.x + threadIdx.x;
  if (i >= NPTS) return;
  uint32_t u = __float_as_uint(conf[i]);
  uint32_t m = (u & 0x80000000u) ? ~u : (u ^ 0x80000000u); // ascending map
  keys[i] = ~m;                                            // descending order
}

__global__ void radix_hist_kernel(const uint32_t* __restrict__ keys, uint32_t shift,
                                  uint32_t* __restrict__ hist) {
  __shared__ uint32_t cnt[256];
  cnt[threadIdx.x] = 0u;
  __syncthreads();
  uint32_t k = keys[blockIdx.x * 256u + threadIdx.x];
  atomicAdd(&cnt[(k >> shift) & 255u], 1u);
  __syncthreads();
  hist[threadIdx.x * RADIX_BLOCKS + blockIdx.x] = cnt[threadIdx.x];
}

__global__ void radix_scatter_kernel(const uint32_t* __restrict__ src,
                                     uint32_t* __restrict__ dst,
                                     const uint32_t* __restrict__ histExcl,
                                     uint32_t shift) {
  __shared__ uint32_t wd[8 * 256];
  for (int q = 0; q < 8; ++q) wd[q * 256 + threadIdx.x] = 0u;
  __syncthreads();
  uint32_t lane = threadIdx.x & 31u, wave = threadIdx.x >> 5;
  uint32_t k = src[blockIdx.x * 256u + threadIdx.x];
  uint32_t d = (k >> shift) & 255u;
  uint32_t mm = 0xFFFFFFFFu;                     // wave32 digit-match mask
  #pragma unroll
  for (int b = 0; b < 8; ++b) {
    uint32_t bal = (uint32_t)__builtin_amdgcn_ballot_w32(((d >> b) & 1u) != 0u);
    mm &= ((d >> b) & 1u) ? bal : ~bal;
  }
  uint32_t rank = __popc(mm & ((1u << lane) - 1u));
  if (rank == 0u) wd[wave * 256u + d] = (uint32_t)__popc(mm);
  __syncthreads();
  uint32_t off = 0u;
  for (uint32_t w = 0; w < wave; ++w) off += wd[w * 256u + d];
  dst[histExcl[d * RADIX_BLOCKS + blockIdx.x] + off + rank] = k;  // stable
}

// ---------------- pass 4: write all outputs ----------------
__global__ void write_out_kernel(const uint32_t* __restrict__ offsets,
                                 const uint32_t* __restrict__ keptOff,
                                 const uint32_t* __restrict__ bPid,
                                 const uint32_t* __restrict__ sortedKeys,
                                 const float* __restrict__ pts,
                                 const float* __restrict__ desc,
                                 const float* __restrict__ desc_conf,
                                 const float* __restrict__ desc_mean,
                                 const float* __restrict__ desc_std,
                                 const float* __restrict__ dc_mean,
                                 const float* __restrict__ dc_std,
                                 const int* __restrict__ voxcnt,
                                 float* __restrict__ fout,
                                 int* __restrict__ grid) {
  uint32_t t = blockIdx.x * blockDim.x + threadIdx.x;
  if (t >= GRID_ELEMS) return;
  uint32_t v = t >> 4, j = t & 15u;
  if ((int)j >= voxcnt[v]) return;               // grid stays -1 from init
  uint32_t pid = bPid[offsets[v] + j];
  uint32_t o = keptOff[v] + j;                   // compact output slot
  grid[v * KCAP + j] = (int)o;
  __builtin_prefetch(&desc[pid * (uint32_t)D_DESC], 0, 0);  // global_prefetch_b8
  fout[PTS_OFF + 3u * o + 0u] = pts[3u * pid + 0u];
  fout[PTS_OFF + 3u * o + 1u] = pts[3u * pid + 1u];
  fout[PTS_OFF + 3u * o + 2u] = pts[3u * pid + 2u];
  // faithful bug: conf_out = conf_sorted[pid] == pid-th largest conf value
  uint32_t m = ~sortedKeys[pid];
  uint32_t cu = (m & 0x80000000u) ? (m ^ 0x80000000u) : ~m;
  fout[CONF_OFF + o] = __uint_as_float(cu);
  fout[DC_OFF + o] = (desc_conf[pid] - dc_mean[0]) / dc_std[0];
  #pragma unroll
  for (int d = 0; d < D_DESC; ++d)
    fout[DESC_OFF + o * (uint32_t)D_DESC + (uint32_t)d] =
        (desc[pid * (uint32_t)D_DESC + (uint32_t)d] - desc_mean[d]) / desc_std[d];
}

// ---------------- host orchestration ----------------
extern "C" void kernel_launch(void* const* d_in, const int* in_sizes, int n_in,
                              void* d_out, int out_size, void* d_ws, size_t ws_size,
                              hipStream_t stream) {
  (void)in_sizes; (void)n_in; (void)out_size; (void)ws_size;
  const float* pts       = (const float*)d_in[0];
  const float* conf      = (const float*)d_in[1];
  const float* desc      = (const float*)d_in[2];
  const float* desc_conf = (const float*)d_in[3];
  const float* desc_mean = (const float*)d_in[4];
  const float* desc_std  = (const float*)d_in[5];
  const float* dc_mean   = (const float*)d_in[6];
  const float* dc_std    = (const float*)d_in[7];
  const float* center    = (const float*)d_in[8];

  float* fout = (float*)d_out;
  int* grid    = (int*)(fout + FLOATS_END);
  int* voxcnt  = grid + GRID_ELEMS;

  // workspace carve-up (256B aligned); keys0/keys1 alias vidArr/bConf (dead by then)
  uint8_t* w = (uint8_t*)d_ws;
  size_t p = 0;
  auto take = [&](size_t bytes) { uint8_t* r = w + p; p += (bytes + 255) & ~(size_t)255; return r; };
  uint32_t* counts   = (uint32_t*)take(COUNTS_PAD * 4);  // tile-padded for scan reads
  uint32_t* offsets  = (uint32_t*)take(NV1 * 4);
  uint32_t* cursor   = (uint32_t*)take(NV1 * 4);
  uint32_t* keptOff  = (uint32_t*)take(NUM_VOX * 4);
  uint32_t* bPid     = (uint32_t*)take(NPTS * 4);
  uint32_t* bConf_ks = (uint32_t*)take(NPTS * 4);   // bConf, later keys ping
  uint32_t* keys0    = (uint32_t*)take(NPTS * 4);   // vidArr, later keys pong
  uint32_t* hist     = (uint32_t*)take(256u * RADIX_BLOCKS * 4);
  uint32_t* partials = (uint32_t*)take(1024 * 4);
  uint32_t* vidArr = keys0;
  float*    bConf  = (float*)bConf_ks;

  // 1) init outputs (zeros + grid=-1) and counts=0 (full padded length)
  init_out_kernel<<<4096, 256, 0, stream>>>((uint32_t*)d_out);
  zero_u32_kernel<<<(COUNTS_PAD + 255u) / 256u, 256, 0, stream>>>(counts, COUNTS_PAD);

  // 2) voxel id + per-voxel counts
  vid_count_kernel<<<NPTS / 256u, 256, 0, stream>>>(pts, center, vidArr, counts);

  // 3) exclusive scan counts -> offsets (and cursor copy) [WMMA scan]
  run_scan(counts, NV1, offsets, cursor, partials, stream);

  // 4) scatter points into voxel buckets
  scatter_kernel<<<NPTS / 256u, 256, 0, stream>>>(vidArr, conf, cursor, bPid, bConf);
  // vidArr dead from here -> keys0 reuse; bConf dead after select -> keys ping

  // 5) per-voxel top-K ordering + vox_counts
  select_kernel<<<NUM_VOX / 256u, 256, 0, stream>>>(counts, offsets, bPid, bConf, voxcnt);

  // 6) exclusive scan kept-counts -> compact offsets [WMMA scan]
  //    (NUM_VOX == 512 * SCAN_TILE exactly; no padding needed)
  run_scan((const uint32_t*)voxcnt, NUM_VOX, keptOff, nullptr, partials, stream);

  // 7) descending value-sort of conf (4-pass stable LSD radix) [WMMA scan on hists]
  //    (hist length == 768 * SCAN_TILE exactly)
  make_keys_kernel<<<RADIX_BLOCKS, 256, 0, stream>>>(conf, keys0);
  uint32_t* src = keys0;
  uint32_t* dst = bConf_ks;
  for (uint32_t pass = 0; pass < 4; ++pass) {
    uint32_t shift = pass * 8u;
    radix_hist_kernel<<<RADIX_BLOCKS, 256, 0, stream>>>(src, shift, hist);
    run_scan(hist, 256u * RADIX_BLOCKS, hist, nullptr, partials, stream);
    radix_scatter_kernel<<<RADIX_BLOCKS, 256, 0, stream>>>(src, dst, hist, shift);
    uint32_t* tmp = src; src = dst; dst = tmp;
  }
  // after 4 passes (even), sorted keys are back in keys0 (== src)

  // 8) gather + scatter all outputs
  write_out_kernel<<<GRID_ELEMS / 256u, 256, 0, stream>>>(
      offsets, keptOff, bPid, src, pts, desc, desc_conf,
      desc_mean, desc_std, dc_mean, dc_std, voxcnt, fout, grid);
}